// GCNLinkPredictor_75539884802811
// MI455X (gfx1250) — compile-verified
//
#include <hip/hip_runtime.h>

typedef __attribute__((ext_vector_type(2))) float v2f;
typedef __attribute__((ext_vector_type(4))) float v4f;
typedef __attribute__((ext_vector_type(8))) float v8f;

#define DF 128  // feature dim (in == hidden == 128)

// ---------------- degree / normalization ----------------

__global__ void deg_init_kernel(float* deg, int n) {
    int i = blockIdx.x * blockDim.x + threadIdx.x;
    if (i < n) deg[i] = 1.0f;  // self loop
}

__global__ void deg_accum_kernel(const int* __restrict__ dst, float* deg, int e) {
    int i = blockIdx.x * blockDim.x + threadIdx.x;
    if (i < e) unsafeAtomicAdd(&deg[dst[i]], 1.0f);
}

__global__ void deg_rsqrt_kernel(float* deg, int n) {
    int i = blockIdx.x * blockDim.x + threadIdx.x;
    if (i < n) deg[i] = rsqrtf(deg[i]);
}

// ---------------- fp32 WMMA GEMM: H[n x 128] = X[n x 128] @ W[128 x 128] ----------------
// Block = 128 threads = 4 waves. Each wave computes a 16-row x 128-col strip.
// W is staged in LDS with K-pairs interleaved: sW[(k/2)*256 + n*2 + (k&1)],
// so a B-fragment (two K values for one column) is one aligned ds_load_b64.

__global__ __launch_bounds__(128) void gemm_wmma_kernel(const float* __restrict__ X,
                                                        const float* __restrict__ W,
                                                        float* __restrict__ H, int nrows) {
    __shared__ float sW[DF * DF];  // 64 KB, interleaved layout
    for (int i = threadIdx.x; i < DF * DF; i += blockDim.x) {
        int k = i >> 7;    // row of W
        int nn = i & 127;  // col of W
        sW[(k >> 1) * 256 + nn * 2 + (k & 1)] = W[i];
    }
    __syncthreads();

    const int wave = threadIdx.x >> 5;
    const int lane = threadIdx.x & 31;
    const int tile = blockIdx.x * 4 + wave;  // 16-row tile index
    const int m0 = tile * 16;
    if (m0 >= nrows) return;  // wave-uniform exit; EXEC stays all-1s for WMMA

    const int half = lane >> 4;  // 0: K=kb,kb+1 ; 1: K=kb+2,kb+3
    const int l = lane & 15;

    v8f acc[8] = {};
    const float* xrow = X + (size_t)(m0 + l) * DF;

    for (int kb = 0; kb < DF; kb += 4) {
        // A fragment: one aligned 8B load (k0 = kb + 2*half)
        v2f a = *(const v2f*)(xrow + kb + half * 2);
        const float* wp = sW + ((kb >> 1) + half) * 256;  // pair row for this half
#pragma unroll
        for (int nt = 0; nt < 8; nt++) {
            // B fragment: one aligned ds_load_b64
            v2f b = *(const v2f*)(wp + (nt * 16 + l) * 2);
            acc[nt] = __builtin_amdgcn_wmma_f32_16x16x4_f32(
                false, a, false, b, (short)0, acc[nt], false, false);
        }
    }

    // C/D layout: VGPR r -> row (m0 + r + 8*half), lane l -> col (nt*16 + l)
#pragma unroll
    for (int nt = 0; nt < 8; nt++) {
        const int col = nt * 16 + l;
#pragma unroll
        for (int r = 0; r < 8; r++) {
            const int row = m0 + r + 8 * half;
            H[(size_t)row * DF + col] = acc[nt][r];
        }
    }
}

// ---------------- aggregation ----------------

// out[i,:] = H[i,:] * dinv[i]^2   (the self-loop term; also initializes accumulator)
__global__ void selfloop_init_kernel(const float* __restrict__ H, const float* __restrict__ dinv,
                                     float* __restrict__ out, int n) {
    int i = blockIdx.x * blockDim.x + threadIdx.x;  // float4 index
    int total = n * (DF / 4);
    if (i >= total) return;
    int node = i >> 5;  // /(128/4)
    float s = dinv[node];
    s = s * s;
    v4f v = ((const v4f*)H)[i];
    ((v4f*)out)[i] = v * s;
}

// wave-per-edge: out[dst,:] += H[src,:] * dinv[src] * dinv[dst]
__global__ __launch_bounds__(256) void scatter_edges_kernel(const int* __restrict__ src,
                                                            const int* __restrict__ dst,
                                                            const float* __restrict__ dinv,
                                                            const float* __restrict__ H,
                                                            float* __restrict__ out, int e) {
    const int lane = threadIdx.x & 31;
    const int w = (int)((blockIdx.x * blockDim.x + threadIdx.x) >> 5);
    if (w >= e) return;
    const int s = src[w];
    const int d = dst[w];
    const float norm = dinv[s] * dinv[d];
    v4f m = ((const v4f*)(H + (size_t)s * DF))[lane];
    float* o = out + (size_t)d * DF + lane * 4;
    unsafeAtomicAdd(o + 0, m.x * norm);
    unsafeAtomicAdd(o + 1, m.y * norm);
    unsafeAtomicAdd(o + 2, m.z * norm);
    unsafeAtomicAdd(o + 3, m.w * norm);
}

// z = relu(agg + bias) or (agg + bias)
__global__ void bias_act_kernel(const float* __restrict__ agg, const float* __restrict__ bias,
                                float* __restrict__ z, int n, int do_relu) {
    int i = blockIdx.x * blockDim.x + threadIdx.x;  // float4 index
    int total = n * (DF / 4);
    if (i >= total) return;
    int c4 = i & 31;
    v4f b = ((const v4f*)bias)[c4];
    v4f v = ((const v4f*)agg)[i] + b;
    if (do_relu) {
        v.x = fmaxf(v.x, 0.0f);
        v.y = fmaxf(v.y, 0.0f);
        v.z = fmaxf(v.z, 0.0f);
        v.w = fmaxf(v.w, 0.0f);
    }
    ((v4f*)z)[i] = v;
}

// ---------------- decode: per-edge dot product ----------------

__global__ __launch_bounds__(256) void decode_kernel(const int* __restrict__ s,
                                                     const int* __restrict__ d,
                                                     const float* __restrict__ z,
                                                     float* __restrict__ logits, int e) {
    const int lane = threadIdx.x & 31;
    const int w = (int)((blockIdx.x * blockDim.x + threadIdx.x) >> 5);
    if (w >= e) return;
    const int si = s[w];
    const int di = d[w];
    v4f a = ((const v4f*)(z + (size_t)si * DF))[lane];
    v4f b = ((const v4f*)(z + (size_t)di * DF))[lane];
    float dot = a.x * b.x + a.y * b.y + a.z * b.z + a.w * b.w;
#pragma unroll
    for (int off = 16; off > 0; off >>= 1) dot += __shfl_xor(dot, off, 32);
    if (lane == 0) logits[w] = dot;
}

// ---------------- launch ----------------

extern "C" void kernel_launch(void* const* d_in, const int* in_sizes, int n_in,
                              void* d_out, int out_size, void* d_ws, size_t ws_size,
                              hipStream_t stream) {
    const float* x  = (const float*)d_in[0];
    const int*   ei = (const int*)d_in[1];   // [2, E] row-major: src then dst
    const int*   el = (const int*)d_in[2];   // [2, EL]
    const float* W1 = (const float*)d_in[3];
    const float* b1 = (const float*)d_in[4];
    const float* W2 = (const float*)d_in[5];
    const float* b2 = (const float*)d_in[6];

    const int n  = in_sizes[0] / DF;   // 100000
    const int E  = in_sizes[1] / 2;    // 1.6M
    const int EL = in_sizes[2] / 2;    // 200K

    float* bufA = (float*)d_ws;                      // n*128 floats
    float* bufB = bufA + (size_t)n * DF;             // n*128 floats
    float* dinv = bufB + (size_t)n * DF;             // n floats
    float* out  = (float*)d_out;

    const int tiles  = (n + 15) / 16;
    const int gemmBl = (tiles + 3) / 4;
    const int elem4  = n * (DF / 4);
    const int elBl   = (elem4 + 255) / 256;
    const int nBl    = (n + 255) / 256;
    const int eBl    = (E + 255) / 256;
    const int scatBl = (E + 7) / 8;    // 8 waves per 256-thread block
    const int decBl  = (EL + 7) / 8;

    // degrees -> dinv (shared by both layers)
    deg_init_kernel<<<nBl, 256, 0, stream>>>(dinv, n);
    deg_accum_kernel<<<eBl, 256, 0, stream>>>(ei + E, dinv, E);
    deg_rsqrt_kernel<<<nBl, 256, 0, stream>>>(dinv, n);

    // layer 1: h1 = x@W1 ; agg ; z1 = relu(agg + b1)
    gemm_wmma_kernel<<<gemmBl, 128, 0, stream>>>(x, W1, bufA, n);
    selfloop_init_kernel<<<elBl, 256, 0, stream>>>(bufA, dinv, bufB, n);
    scatter_edges_kernel<<<scatBl, 256, 0, stream>>>(ei, ei + E, dinv, bufA, bufB, E);
    bias_act_kernel<<<elBl, 256, 0, stream>>>(bufB, b1, bufA, n, 1);

    // layer 2: h2 = z1@W2 ; agg ; z2 = agg + b2
    gemm_wmma_kernel<<<gemmBl, 128, 0, stream>>>(bufA, W2, bufB, n);
    selfloop_init_kernel<<<elBl, 256, 0, stream>>>(bufB, dinv, bufA, n);
    scatter_edges_kernel<<<scatBl, 256, 0, stream>>>(ei, ei + E, dinv, bufB, bufA, E);
    bias_act_kernel<<<elBl, 256, 0, stream>>>(bufA, b2, bufB, n, 0);

    // decode
    decode_kernel<<<decBl, 256, 0, stream>>>(el, el + EL, bufB, out, EL);
}